// ExplicitGCN_90829968375999
// MI455X (gfx1250) — compile-verified
//
#include <hip/hip_runtime.h>

// ---------------- problem constants (match reference) ----------------
#define V_   50000
#define E_   300000
#define B_   4
#define LAT_ 512
#define H_   128
#define L_   3
#define N_   (B_ * V_)          // 200000 flattened nodes

typedef float v2f __attribute__((ext_vector_type(2)));
typedef float v8f __attribute__((ext_vector_type(8)));

// ---------------------------------------------------------------------
// deg[c] += 1 per original edge (col side); self-loops/B-tiling folded later
__global__ __launch_bounds__(256) void deg_kernel(const int* __restrict__ ei,
                                                  int* __restrict__ deg) {
    int e = blockIdx.x * 256 + threadIdx.x;
    if (e < E_) atomicAdd(&deg[ei[E_ + e]], 1);   // ei[1][e] = target
}

// dis[v] = rsqrt(B*indeg(v) + 1)   (nodes >= V have deg 1 -> dis == 1, implicit)
__global__ __launch_bounds__(256) void dis_kernel(const int* __restrict__ deg,
                                                  float* __restrict__ dis) {
    int v = blockIdx.x * 256 + threadIdx.x;
    if (v < V_) dis[v] = rsqrtf((float)B_ * (float)deg[v] + 1.0f);
}

// latproj[b][h] = b_in[h] + sum_k latent[b][k] * W_in[h][3+k]
__global__ __launch_bounds__(256) void latproj_kernel(const float* __restrict__ latent,
                                                      const float* __restrict__ W_in,
                                                      const float* __restrict__ b_in,
                                                      float* __restrict__ latproj) {
    int t = blockIdx.x * 256 + threadIdx.x;
    if (t >= B_ * H_) return;
    int b = t >> 7, h = t & (H_ - 1);
    const float* w   = W_in + (size_t)h * (LAT_ + 3) + 3;
    const float* lat = latent + (size_t)b * LAT_;
    float s = b_in[h];
    for (int k = 0; k < LAT_; ++k) s = fmaf(lat[k], w[k], s);
    latproj[t] = s;
}

// X[n][h] = relu( xyz[v]·W_in[h][0:3] + latproj[b][h] ),  n = b*V + v
__global__ __launch_bounds__(256) void input_kernel(const float* __restrict__ xyz,
                                                    const float* __restrict__ W_in,
                                                    const float* __restrict__ latproj,
                                                    float* __restrict__ X) {
    size_t gid = (size_t)blockIdx.x * 256 + threadIdx.x;
    if (gid >= (size_t)N_ * H_) return;
    int h = (int)(gid & (H_ - 1));
    int n = (int)(gid >> 7);
    int v = n % V_;
    int b = n / V_;
    const float* w = W_in + (size_t)h * (LAT_ + 3);
    float s = latproj[b * H_ + h];
    s = fmaf(xyz[v * 3 + 0], w[0], s);
    s = fmaf(xyz[v * 3 + 1], w[1], s);
    s = fmaf(xyz[v * 3 + 2], w[2], s);
    X[gid] = fmaxf(s, 0.0f);
}

// ---------------------------------------------------------------------
// H = X @ Wl^T  via V_WMMA_F32_16X16X4_F32; epilogue also writes the
// self-loop-initialized aggregator: Agg = (r<V ? dis[r]^2 : 1) * H.
// Block: 256 threads = 8 waves; block covers 128 rows x 128 cols.
// Wave w owns rows [w*16, w*16+16), all 8 column tiles (8x v8f accumulators).
//
// LDS holds W in k-paired transposed form:
//   sWp[ ((k>>1)*H + j)*2 + (k&1) ] = W[j][k]
// so a B fragment (two consecutive k for one j) is ONE aligned ds_load_b64,
// and lane l reads banks {2l, 2l+1} -> conflict-free.
__global__ __launch_bounds__(256) void gemm_kernel(const float* __restrict__ X,
                                                   const float* __restrict__ W,
                                                   float* __restrict__ Hout,
                                                   float* __restrict__ Agg,
                                                   const float* __restrict__ dis) {
    __shared__ float sWp[H_ * H_];     // 64 KB

    const int t = threadIdx.x;
    // Coalesced float4 reads of W (row-major [j][k]); paired-transposed LDS writes.
    for (int i = t; i < (H_ * H_) / 4; i += 256) {
        const int j = (i * 4) >> 7;
        const int k = (i * 4) & (H_ - 1);          // multiple of 4
        const float4 w = reinterpret_cast<const float4*>(W)[i];
        v2f lo; lo.x = w.x; lo.y = w.y;            // k, k+1
        v2f hi; hi.x = w.z; hi.y = w.w;            // k+2, k+3
        const int kp = k >> 1;
        *reinterpret_cast<v2f*>(&sWp[(kp * H_ + j) * 2])       = lo;
        *reinterpret_cast<v2f*>(&sWp[((kp + 1) * H_ + j) * 2]) = hi;
    }
    __syncthreads();

    const int wave   = t >> 5;
    const int lane   = t & 31;
    const int laneLo = lane & 15;
    const int khalf  = (lane >> 4) << 1;              // 0 for lanes 0-15, 2 for 16-31
    const int rbase  = blockIdx.x * 128 + wave * 16;  // tile row base
    const int arow   = rbase + laneLo;                // A-fragment row fed by this lane
    // Clamp instead of predicating: garbage A row m only affects D row m,
    // and OOB D rows are skipped at store time. Keeps the main loop branch-free.
    const int arowC  = arow < N_ ? arow : (N_ - 1);
    const float* xrow = X + (size_t)arowC * H_;

    v8f acc[8] = {};

    for (int k0 = 0; k0 < H_; k0 += 4) {
        // A fragment (16x4 fp32): lane<16 holds K=k0,k0+1; lane>=16 holds K=k0+2,k0+3
        const v2f a = *reinterpret_cast<const v2f*>(xrow + k0 + khalf);
        const int kp = (k0 + khalf) >> 1;
        const float* wrow = &sWp[(kp * H_ + laneLo) * 2];
#pragma unroll
        for (int ct = 0; ct < 8; ++ct) {
            // B fragment (4x16 fp32): B[k][j] = W[j][k]; one aligned b64 LDS read.
            const v2f b = *reinterpret_cast<const v2f*>(wrow + ct * 32);
            acc[ct] = __builtin_amdgcn_wmma_f32_16x16x4_f32(
                false, a, false, b, (short)0, acc[ct], false, false);
        }
    }

    // D layout: VGPR i -> row M = i + (lane>=16 ? 8 : 0), col N = laneLo
    const int rhalf = (lane >> 4) * 8;
    float scl[8];
#pragma unroll
    for (int i = 0; i < 8; ++i) {
        const int r = rbase + rhalf + i;
        if (r < V_) { const float d = dis[r]; scl[i] = d * d; }
        else        { scl[i] = 1.0f; }
    }
#pragma unroll
    for (int ct = 0; ct < 8; ++ct) {
        const int col = ct * 16 + laneLo;
#pragma unroll
        for (int i = 0; i < 8; ++i) {
            const int r = rbase + rhalf + i;
            if (r < N_) {
                const float v = acc[ct][i];
                Hout[(size_t)r * H_ + col] = v;
                Agg[(size_t)r * H_ + col] = scl[i] * v;   // self-loop term
            }
        }
    }
}

// One wave per edge: Agg[c] += B * dis[r]*dis[c] * H[r]  (128 floats, 4/lane)
__global__ __launch_bounds__(256) void edge_kernel(const int* __restrict__ ei,
                                                   const float* __restrict__ Hbuf,
                                                   float* __restrict__ Agg,
                                                   const float* __restrict__ dis) {
    const size_t gid = (size_t)blockIdx.x * 256 + threadIdx.x;
    const int edge = (int)(gid >> 5);
    const int lane = (int)(gid & 31);
    if (edge >= E_) return;
    const int r = ei[edge];        // source (row)
    const int c = ei[E_ + edge];   // target (col)
    const float w = (float)B_ * dis[r] * dis[c];
    const float4 v = reinterpret_cast<const float4*>(Hbuf + (size_t)r * H_)[lane];
    float* a = Agg + (size_t)c * H_ + lane * 4;
    atomicAdd(a + 0, w * v.x);
    atomicAdd(a + 1, w * v.y);
    atomicAdd(a + 2, w * v.z);
    atomicAdd(a + 3, w * v.w);
}

// X = relu(Agg + conv_b[l])
__global__ __launch_bounds__(256) void relu_bias_kernel(const float* __restrict__ Agg,
                                                        const float* __restrict__ bias,
                                                        float* __restrict__ X) {
    size_t gid = (size_t)blockIdx.x * 256 + threadIdx.x;
    if (gid >= (size_t)N_ * H_) return;
    X[gid] = fmaxf(Agg[gid] + bias[gid & (H_ - 1)], 0.0f);
}

// out[n] = x[n] @ W_out^T + b_out   (N x 3)
__global__ __launch_bounds__(256) void out_kernel(const float* __restrict__ X,
                                                  const float* __restrict__ W_out,
                                                  const float* __restrict__ b_out,
                                                  float* __restrict__ out) {
    int n = blockIdx.x * 256 + threadIdx.x;
    if (n >= N_) return;
    const float* x = X + (size_t)n * H_;
    float s0 = b_out[0], s1 = b_out[1], s2 = b_out[2];
    for (int k = 0; k < H_; ++k) {
        const float xv = x[k];
        s0 = fmaf(xv, W_out[k], s0);
        s1 = fmaf(xv, W_out[H_ + k], s1);
        s2 = fmaf(xv, W_out[2 * H_ + k], s2);
    }
    out[(size_t)n * 3 + 0] = s0;
    out[(size_t)n * 3 + 1] = s1;
    out[(size_t)n * 3 + 2] = s2;
}

// ---------------------------------------------------------------------
extern "C" void kernel_launch(void* const* d_in, const int* in_sizes, int n_in,
                              void* d_out, int out_size, void* d_ws, size_t ws_size,
                              hipStream_t stream) {
    const float* xyz   = (const float*)d_in[0];
    const float* lat   = (const float*)d_in[1];
    const int*   ei    = (const int*)d_in[2];
    const float* W_in  = (const float*)d_in[3];
    const float* b_in  = (const float*)d_in[4];
    const float* convW = (const float*)d_in[5];
    const float* convb = (const float*)d_in[6];
    const float* W_out = (const float*)d_in[7];
    const float* b_out = (const float*)d_in[8];
    float* out = (float*)d_out;

    char* ws = (char*)d_ws;
    auto aln = [](size_t x) { return (x + 255) & ~(size_t)255; };
    size_t off = 0;
    float* X    = (float*)(ws + off); off = aln(off + (size_t)N_ * H_ * 4);
    float* Hbuf = (float*)(ws + off); off = aln(off + (size_t)N_ * H_ * 4);
    float* Agg  = (float*)(ws + off); off = aln(off + (size_t)N_ * H_ * 4);
    float* dis  = (float*)(ws + off); off = aln(off + (size_t)V_ * 4);
    int*   deg  = (int*)  (ws + off); off = aln(off + (size_t)V_ * 4);
    float* lpj  = (float*)(ws + off); off = aln(off + (size_t)B_ * H_ * 4);

    hipMemsetAsync(deg, 0, (size_t)V_ * sizeof(int), stream);
    deg_kernel<<<(E_ + 255) / 256, 256, 0, stream>>>(ei, deg);
    dis_kernel<<<(V_ + 255) / 256, 256, 0, stream>>>(deg, dis);
    latproj_kernel<<<(B_ * H_ + 255) / 256, 256, 0, stream>>>(lat, W_in, b_in, lpj);
    {
        size_t tot = (size_t)N_ * H_;
        input_kernel<<<(unsigned)((tot + 255) / 256), 256, 0, stream>>>(xyz, W_in, lpj, X);
    }

    for (int l = 0; l < L_; ++l) {
        gemm_kernel<<<(N_ + 127) / 128, 256, 0, stream>>>(
            X, convW + (size_t)l * H_ * H_, Hbuf, Agg, dis);
        edge_kernel<<<(E_ * 32) / 256, 256, 0, stream>>>(ei, Hbuf, Agg, dis);
        size_t tot = (size_t)N_ * H_;
        relu_bias_kernel<<<(unsigned)((tot + 255) / 256), 256, 0, stream>>>(
            Agg, convb + l * H_, X);
    }
    out_kernel<<<(N_ + 255) / 256, 256, 0, stream>>>(X, W_out, b_out, out);
}